// IoULoss_31765578121756
// MI455X (gfx1250) — compile-verified
//
#include <hip/hip_runtime.h>

#define NUM_CLASSES 21
#define HW (512 * 512)
#define HW4 (HW / 4)            // 65536 float4 groups per plane
#define NBATCH 16
#define NQUAD (NBATCH * HW4)    // 1,048,576 quad-pixels total
#define WAVES_PER_BLOCK 8
#define HIST_STRIDE 96          // per-wave hist: [0..20]=pred, [32..52]=tgt, [64..84]=inter

typedef float v2f __attribute__((ext_vector_type(2)));
typedef float v8f __attribute__((ext_vector_type(8)));

// ---------------------------------------------------------------------------
// Kernel 1: streaming argmax + per-wave LDS histograms -> global counters.
// Memory-bound: 369 MB @ 23.3 TB/s ~= 15.8 us floor. Everything else hides.
// ---------------------------------------------------------------------------
__global__ void __launch_bounds__(256)
iou_hist_kernel(const float* __restrict__ inputs,
                const int*   __restrict__ targets,
                unsigned*    __restrict__ ghist) {
    __shared__ unsigned hist[WAVES_PER_BLOCK * HIST_STRIDE];

    const int wave = threadIdx.x >> 5;            // wave32 on gfx1250
    unsigned* h = &hist[wave * HIST_STRIDE];

    for (int i = threadIdx.x; i < WAVES_PER_BLOCK * HIST_STRIDE; i += blockDim.x)
        hist[i] = 0u;
    __syncthreads();

    const int stride = gridDim.x * blockDim.x;    // in quad-pixels
    const float4* in4 = reinterpret_cast<const float4*>(inputs);
    const int4*   tg4 = reinterpret_cast<const int4*>(targets);

    for (int q = blockIdx.x * blockDim.x + threadIdx.x; q < NQUAD; q += stride) {
        const int b  = q >> 16;            // q / HW4
        const int p4 = q & (HW4 - 1);      // q % HW4
        const float4* base = in4 + (size_t)b * (NUM_CLASSES * HW4) + p4;

        // Prefetch next grid-stride tile (speculative; OOB is silently dropped).
        __builtin_prefetch(tg4 + q + stride, 0, 0);
        __builtin_prefetch(base + stride, 0, 0);

        float4 m = base[0];
        int ax = 0, ay = 0, az = 0, aw = 0;
#pragma unroll
        for (int c = 1; c < NUM_CLASSES; ++c) {
            __builtin_prefetch(base + (size_t)c * HW4 + stride, 0, 0);
            float4 v = base[(size_t)c * HW4];
            if (v.x > m.x) { m.x = v.x; ax = c; }   // strict > => first max wins,
            if (v.y > m.y) { m.y = v.y; ay = c; }   // matching jnp.argmax
            if (v.z > m.z) { m.z = v.z; az = c; }
            if (v.w > m.w) { m.w = v.w; aw = c; }
        }

        const int4 t = tg4[q];

        atomicAdd(&h[ax], 1u);
        atomicAdd(&h[ay], 1u);
        atomicAdd(&h[az], 1u);
        atomicAdd(&h[aw], 1u);
        atomicAdd(&h[32 + t.x], 1u);
        atomicAdd(&h[32 + t.y], 1u);
        atomicAdd(&h[32 + t.z], 1u);
        atomicAdd(&h[32 + t.w], 1u);
        if (ax == t.x) atomicAdd(&h[64 + ax], 1u);
        if (ay == t.y) atomicAdd(&h[64 + ay], 1u);
        if (az == t.z) atomicAdd(&h[64 + az], 1u);
        if (aw == t.w) atomicAdd(&h[64 + aw], 1u);
    }
    __syncthreads();

    // Flush block's 63 counters to global (padded layout identical to LDS).
    for (int i = threadIdx.x; i < 3 * NUM_CLASSES; i += blockDim.x) {
        const int arr = i / NUM_CLASSES;
        const int bin = i % NUM_CLASSES;
        unsigned s = 0;
#pragma unroll
        for (int w = 0; w < WAVES_PER_BLOCK; ++w)
            s += hist[w * HIST_STRIDE + arr * 32 + bin];
        atomicAdd(&ghist[arr * 32 + bin], s);
    }
}

// ---------------------------------------------------------------------------
// Kernel 2: one wave32. Per-class IoU losses in f32, summed exactly via
// V_WMMA_F32_16X16X4_F32 (A = losses, B = ones => D rows are row-sums),
// then one shfl_xor(16) to combine the two half-wave partials.
// EXEC is all-1s at the WMMA (uniform control flow, 32 threads launched).
// ---------------------------------------------------------------------------
__global__ void __launch_bounds__(32)
iou_finalize_kernel(const unsigned* __restrict__ gh, float* __restrict__ out) {
    const int lane = threadIdx.x;

    v2f a = {0.0f, 0.0f};
    if (lane < 16) {
        {
            const int c = lane;                         // classes 0..15 -> A[m][K=0]
            const float I = (float)gh[64 + c];
            const float P = (float)gh[c];
            const float T = (float)gh[32 + c];
            const float U = P + T - I;
            a.x = 1.0f - (I + 1.0f) / (U + 1.0f);
        }
        if (lane < 5) {
            const int c = 16 + lane;                    // classes 16..20 -> A[m][K=1]
            const float I = (float)gh[64 + c];
            const float P = (float)gh[c];
            const float T = (float)gh[32 + c];
            const float U = P + T - I;
            a.y = 1.0f - (I + 1.0f) / (U + 1.0f);
        }
    }
    // lanes 16..31 carry K=2,3 of A: left zero.

    v2f bv = {1.0f, 1.0f};   // B = all-ones 4x16 (layout-agnostic)
    v8f cacc = {};
    v8f d = __builtin_amdgcn_wmma_f32_16x16x4_f32(
        /*neg_a=*/false, a, /*neg_b=*/false, bv,
        /*c_mod=*/(short)0, cacc, /*reuse_a=*/false, /*reuse_b=*/false);

    // Each lane's 8 VGPRs hold D[m][n] for 8 distinct m; columns are identical.
    float partial = d[0] + d[1] + d[2] + d[3] + d[4] + d[5] + d[6] + d[7];
    partial += __shfl_xor(partial, 16, 32);

    if (lane == 0)
        out[0] = partial * (1.0f / (float)(NUM_CLASSES - 1));
}

// ---------------------------------------------------------------------------
extern "C" void kernel_launch(void* const* d_in, const int* in_sizes, int n_in,
                              void* d_out, int out_size, void* d_ws, size_t ws_size,
                              hipStream_t stream) {
    (void)in_sizes; (void)n_in; (void)out_size; (void)ws_size;

    const float* inputs  = (const float*)d_in[0];   // [16, 21, 512, 512] f32
    const int*   targets = (const int*)d_in[1];     // [16, 512, 512] i32
    float*       out     = (float*)d_out;           // scalar f32
    unsigned*    ghist   = (unsigned*)d_ws;         // 96 u32 counters (padded)

    hipMemsetAsync(ghist, 0, 96 * sizeof(unsigned), stream);   // capture-safe

    iou_hist_kernel<<<2048, 256, 0, stream>>>(inputs, targets, ghist);
    iou_finalize_kernel<<<1, 32, 0, stream>>>(ghist, out);
}